// GAT_80135499809441
// MI455X (gfx1250) — compile-verified
//
#include <hip/hip_runtime.h>
#include <hip/hip_bf16.h>

typedef __attribute__((ext_vector_type(2))) float v2f;
typedef __attribute__((ext_vector_type(8))) float v8f;

// ---------------- helpers: ordered-float encoding for atomic max ----------------
__device__ __forceinline__ unsigned f2ord(float f) {
  unsigned u = __float_as_uint(f);
  return (u & 0x80000000u) ? ~u : (u | 0x80000000u);
}
__device__ __forceinline__ float ord2f(unsigned u) {
  return (u & 0x80000000u) ? __uint_as_float(u & 0x7FFFFFFFu) : __uint_as_float(~u);
}
#define ORD_NEG_INF 0x007FFFFFu  // f2ord(-inf)

// ---------------- generic fill ----------------
__global__ void fill_u32(unsigned* __restrict__ p, unsigned val, int n) {
  int t = blockIdx.x * blockDim.x + threadIdx.x;
  if (t < n) p[t] = val;
}

// ---------------- prep: fold fe-expansion into 32-vectors ----------------
// cvec layout: c1[32] c2[32] c3[32] ce[32]
__global__ void prep_mlp(const float* __restrict__ W1c, const float* __restrict__ W1t,
                         const float* __restrict__ W1e, float* __restrict__ cvec) {
  int k = threadIdx.x;
  if (k >= 32) return;
  float s1 = 0.f, s2 = 0.f, s3 = 0.f, s4 = 0.f;
  for (int j = 0; j < 64; ++j) {
    float fe = (float)j / 63.f; fe *= fe;
    s1 += fe * W1c[j * 32 + k];
    s2 += fe * W1c[(64 + j) * 32 + k];
    s3 += fe * W1t[j * 32 + k];
    s4 += fe * W1e[j * 32 + k];
  }
  cvec[k] = s1; cvec[32 + k] = s2; cvec[64 + k] = s3; cvec[96 + k] = s4;
}

// ---------------- prep per conv layer ----------------
// Waug: (din x 208) row-major = [Wsrc(192) | usrc heads(3) | udst heads(3) | zeros(10)]
// v layout: [h*32 + k] ; bsc[h]
#define MAUG 208
__global__ void prep_layer(const float* __restrict__ Wsrc, const float* __restrict__ Wdst,
                           const float* __restrict__ Wedge,
                           const float* __restrict__ attS, const float* __restrict__ attD,
                           const float* __restrict__ attE,
                           const float* __restrict__ W2e, const float* __restrict__ b2e,
                           int din, float* __restrict__ Waug, float* __restrict__ v,
                           float* __restrict__ bsc) {
  __shared__ float we[3][64];
  int t = threadIdx.x;
  if (t < 192) {
    int h_ = t / 64, j = t % 64;
    float s = 0.f;
    for (int c = 0; c < 64; ++c) s += Wedge[j * 192 + h_ * 64 + c] * attE[h_ * 64 + c];
    we[h_][j] = s;
  }
  __syncthreads();
  if (t < 96) {
    int h_ = t / 32, k = t % 32;
    float s = 0.f;
    for (int j = 0; j < 64; ++j) s += W2e[k * 64 + j] * we[h_][j];
    v[h_ * 32 + k] = s;
  } else if (t < 99) {
    int h_ = t - 96;
    float s = 0.f;
    for (int j = 0; j < 64; ++j) s += b2e[j] * we[h_][j];
    bsc[h_] = s;
  }
  // copy Wsrc into Waug cols 0..191
  for (int i = t; i < din * 192; i += 256) {
    int k = i / 192, c = i % 192;
    Waug[k * MAUG + c] = Wsrc[i];
  }
  // attention projection columns 192..197, zero pad 198..207
  for (int i = t; i < din * 16; i += 256) {
    int k = i / 16, r = i % 16;
    float s = 0.f;
    if (r < 3) {
      for (int c = 0; c < 64; ++c) s += Wsrc[k * 192 + r * 64 + c] * attS[r * 64 + c];
    } else if (r < 6) {
      int h_ = r - 3;
      for (int c = 0; c < 64; ++c) s += Wdst[k * 192 + h_ * 64 + c] * attD[h_ * 64 + c];
    }
    Waug[k * MAUG + 192 + r] = s;
  }
}

// ---------------- node MLP: builds h0 (N x 128) ----------------
__global__ void node_mlp(const float* __restrict__ x, const float* __restrict__ xrand,
                         const float* __restrict__ cvec,
                         const float* __restrict__ b1c, const float* __restrict__ W2c,
                         const float* __restrict__ b2c,
                         const float* __restrict__ b1t, const float* __restrict__ W2t,
                         const float* __restrict__ b2t,
                         float* __restrict__ h0, int n) {
  int t = blockIdx.x * blockDim.x + threadIdx.x;
  if (t >= n * 64) return;
  int node = t >> 6, c = t & 63;
  float xm = x[node * 5 + 0];
  float x0 = x[node * 5 + 1], x1 = x[node * 5 + 2], x2 = x[node * 5 + 3];
  float aC = b2c[c], aT = b2t[c];
  for (int k = 0; k < 32; ++k) {
    float zc = fmaf(x0, cvec[k], fmaf(x1, cvec[32 + k], b1c[k]));
    zc = zc > 0.f ? zc : 0.f;
    aC = fmaf(zc, W2c[k * 64 + c], aC);
    float zt = fmaf(x2, cvec[64 + k], b1t[k]);
    zt = zt > 0.f ? zt : 0.f;
    aT = fmaf(zt, W2t[k * 64 + c], aT);
  }
  h0[(size_t)node * 128 + c] = aC * (1.f - xm) + aT * xm;
  h0[(size_t)node * 128 + 64 + c] = xrand[(size_t)node * 64 + c];
}

// ---------------- WMMA GEMM: C(N x MT*16) = A(N x K) @ B(K x MT*16), f32 ----------------
// B staged through LDS swizzled by 4-row k-groups so each fragment is one ds_load_b64:
//   Bs[(kg*M + col)*4 + j]  holds  B[kc + kg*4 + j][col]
template <int K, int MT>
__global__ __launch_bounds__(256) void gemm_wmma(const float* __restrict__ A,
                                                 const float* __restrict__ B,
                                                 float* __restrict__ C, int nrows) {
  constexpr int M = MT * 16;
  constexpr int KC = 64;
  __shared__ float Bs[KC * M];  // 52 KB for MT=13
  const int tid = threadIdx.x;
  const int wave = tid >> 5, lane = tid & 31;
  const int m = lane & 15;
  const int khalf = (lane >> 4) << 1;  // 0 or 2
  const int hi8 = (lane >> 4) << 3;    // 0 or 8
  const int row0 = blockIdx.x * 128 + wave * 16;
  const int row = row0 + m;
  const bool rowok = row < nrows;
  const int rowc = rowok ? row : (nrows - 1);
  const float* arow = A + (size_t)rowc * K;

  v8f acc[MT] = {};
  for (int kc = 0; kc < K; kc += KC) {
    __syncthreads();
    for (int i = tid; i < KC * M; i += 256) {
      int r = i / M, c = i - r * M;
      Bs[((r >> 2) * M + c) * 4 + (r & 3)] = B[(size_t)(kc + r) * M + c];
    }
    __syncthreads();
    for (int kk = 0; kk < KC; kk += 4) {
      v2f a;
      a.x = arow[kc + kk + khalf];
      a.y = arow[kc + kk + khalf + 1];
      a.x = rowok ? a.x : 0.f;
      a.y = rowok ? a.y : 0.f;
      const int kg = kk >> 2;
#pragma unroll
      for (int t = 0; t < MT; ++t) {
        v2f b = *(const v2f*)&Bs[((kg * M) + t * 16 + m) * 4 + khalf];
        acc[t] = __builtin_amdgcn_wmma_f32_16x16x4_f32(false, a, false, b, (short)0,
                                                       acc[t], false, false);
      }
    }
  }
#pragma unroll
  for (int t = 0; t < MT; ++t) {
#pragma unroll
    for (int v = 0; v < 8; ++v) {
      int r = row0 + hi8 + v;
      if (r < nrows) C[(size_t)r * M + t * 16 + m] = acc[t][v];
    }
  }
}

// ---------------- edge logits + segment max ----------------
// hs row = [hs(192) | asrc(3) | adst(3) | pad], stride 208
__global__ void attn_max(const int* __restrict__ src, const int* __restrict__ dst,
                         const float* __restrict__ edge_attr,
                         const float* __restrict__ ce, const float* __restrict__ b1e,
                         const float* __restrict__ v, const float* __restrict__ bsc,
                         const float* __restrict__ hs,
                         float* __restrict__ a_raw, unsigned* __restrict__ amax, int E_) {
  int e = blockIdx.x * blockDim.x + threadIdx.x;
  if (e >= E_) return;
  float ea = edge_attr[e];
  float ae0 = bsc[0], ae1 = bsc[1], ae2 = bsc[2];
  for (int k = 0; k < 32; ++k) {
    float z = fmaf(ea, ce[k], b1e[k]);
    z = z > 0.f ? z : 0.f;
    ae0 = fmaf(z, v[k], ae0);
    ae1 = fmaf(z, v[32 + k], ae1);
    ae2 = fmaf(z, v[64 + k], ae2);
  }
  int s = src[e], d = dst[e];
#pragma unroll
  for (int h_ = 0; h_ < 3; ++h_) {
    float aeh = (h_ == 0) ? ae0 : ((h_ == 1) ? ae1 : ae2);
    float a = hs[(size_t)s * MAUG + 192 + h_] + hs[(size_t)d * MAUG + 195 + h_] + aeh;
    a = a > 0.f ? a : 0.2f * a;  // leaky_relu(0.2)
    a_raw[e * 3 + h_] = a;
    atomicMax(&amax[d * 3 + h_], f2ord(a));
  }
}

// ---------------- exp + denom (in-place a_raw -> w) ----------------
__global__ void attn_weight(const int* __restrict__ dst, float* __restrict__ a_raw,
                            const unsigned* __restrict__ amax, float* __restrict__ denom,
                            int total) {
  int t = blockIdx.x * blockDim.x + threadIdx.x;
  if (t >= total) return;
  int e = t / 3, h_ = t - e * 3;
  int d = dst[e];
  float w = __expf(a_raw[t] - ord2f(amax[d * 3 + h_]));
  a_raw[t] = w;
  atomicAdd(&denom[d * 3 + h_], w);
}

// ---------------- weighted message scatter: 4 channels / thread ----------------
__global__ void scatter_msg4(const int* __restrict__ src, const int* __restrict__ dst,
                             const float* __restrict__ hs, const float* __restrict__ w,
                             float* __restrict__ agg, int total /* E*48 */) {
  int t = blockIdx.x * blockDim.x + threadIdx.x;
  if (t >= total) return;
  unsigned e = (unsigned)t / 48u;
  unsigned c = ((unsigned)t - e * 48u) * 4u;  // 0,4,...,188
  unsigned h_ = c >> 6;
  float wv = w[e * 3u + h_];
  const float4 hv = *(const float4*)(hs + (size_t)src[e] * MAUG + c);
  float* ap = agg + (size_t)dst[e] * 192u + c;
  atomicAdd(ap + 0, wv * hv.x);
  atomicAdd(ap + 1, wv * hv.y);
  atomicAdd(ap + 2, wv * hv.z);
  atomicAdd(ap + 3, wv * hv.w);
}

// ---------------- normalize + bias, in place ----------------
__global__ void finalize_layer(float* __restrict__ agg, const float* __restrict__ denom,
                               const float* __restrict__ bias, int total) {
  int t = blockIdx.x * blockDim.x + threadIdx.x;
  if (t >= total) return;
  unsigned node = (unsigned)t / 192u;
  unsigned c = (unsigned)t - node * 192u;
  unsigned h_ = c >> 6;
  agg[t] = agg[t] / (denom[node * 3u + h_] + 1e-16f) + bias[c];
}

// ---------------- conv3 per-node projections ----------------
__global__ void alpha3_kernel(const float* __restrict__ h2, const float* __restrict__ Wsrc3,
                              const float* __restrict__ Wdst3, const float* __restrict__ attS,
                              const float* __restrict__ attD,
                              float* __restrict__ hs3, float* __restrict__ as3,
                              float* __restrict__ ad3, int n) {
  __shared__ float ss[256], sd[256];
  int node = blockIdx.x;
  if (node >= n) return;
  int t = threadIdx.x;
  float a = 0.f, b = 0.f;
  if (t < 192) {
    float hv = h2[(size_t)node * 192 + t];
    a = hv * Wsrc3[t];
    b = hv * Wdst3[t];
  }
  ss[t] = a; sd[t] = b;
  __syncthreads();
  for (int s_ = 128; s_ > 0; s_ >>= 1) {
    if (t < s_) { ss[t] += ss[t + s_]; sd[t] += sd[t + s_]; }
    __syncthreads();
  }
  if (t == 0) {
    hs3[node] = ss[0];
    as3[node] = ss[0] * attS[0];
    ad3[node] = sd[0] * attD[0];
  }
}

__global__ void attn_max3(const int* __restrict__ src, const int* __restrict__ dst,
                          const float* __restrict__ as3, const float* __restrict__ ad3,
                          float* __restrict__ a_raw, unsigned* __restrict__ amax, int E_) {
  int e = blockIdx.x * blockDim.x + threadIdx.x;
  if (e >= E_) return;
  float a = as3[src[e]] + ad3[dst[e]];
  a = a > 0.f ? a : 0.2f * a;
  a_raw[e] = a;
  atomicMax(&amax[dst[e]], f2ord(a));
}

__global__ void attn_wsum3(const int* __restrict__ src, const int* __restrict__ dst,
                           const float* __restrict__ a_raw, const unsigned* __restrict__ amax,
                           const float* __restrict__ hs3, float* __restrict__ denom,
                           float* __restrict__ agg, int E_) {
  int e = blockIdx.x * blockDim.x + threadIdx.x;
  if (e >= E_) return;
  int d = dst[e];
  float w = __expf(a_raw[e] - ord2f(amax[d]));
  atomicAdd(&denom[d], w);
  atomicAdd(&agg[d], w * hs3[src[e]]);
}

__global__ void final3(const float* __restrict__ agg, const float* __restrict__ denom,
                       const float* __restrict__ bias, const float* __restrict__ x,
                       float* __restrict__ out, int n) {
  int t = blockIdx.x * blockDim.x + threadIdx.x;
  if (t >= n) return;
  out[t] = (agg[t] / (denom[t] + 1e-16f) + bias[0]) * x[t * 5 + 0];
}

// =============================== host dispatch ===============================
extern "C" void kernel_launch(void* const* d_in, const int* in_sizes, int n_in,
                              void* d_out, int out_size, void* d_ws, size_t ws_size,
                              hipStream_t stream) {
  // jax tree-flatten order (dict keys sorted; 'W' < 'a' < 'b' in ASCII)
  const float* edge_attr = (const float*)d_in[0];
  const int*   ei        = (const int*)d_in[1];
  const float* W1c = (const float*)d_in[2];
  const float* W2c = (const float*)d_in[3];
  const float* b1c = (const float*)d_in[4];
  const float* b2c = (const float*)d_in[5];
  const float* Wdst1  = (const float*)d_in[6];
  const float* Wedge1 = (const float*)d_in[7];
  const float* Wsrc1  = (const float*)d_in[8];
  const float* attD1  = (const float*)d_in[9];
  const float* attE1  = (const float*)d_in[10];
  const float* attS1  = (const float*)d_in[11];
  const float* bias1  = (const float*)d_in[12];
  const float* Wdst2  = (const float*)d_in[13];
  const float* Wedge2 = (const float*)d_in[14];
  const float* Wsrc2  = (const float*)d_in[15];
  const float* attD2  = (const float*)d_in[16];
  const float* attE2  = (const float*)d_in[17];
  const float* attS2  = (const float*)d_in[18];
  const float* bias2  = (const float*)d_in[19];
  const float* Wdst3  = (const float*)d_in[20];
  const float* Wsrc3  = (const float*)d_in[21];
  const float* attD3  = (const float*)d_in[22];
  const float* attS3  = (const float*)d_in[23];
  const float* bias3  = (const float*)d_in[24];
  const float* W1e = (const float*)d_in[25];
  const float* W2e = (const float*)d_in[26];
  const float* b1e = (const float*)d_in[27];
  const float* b2e = (const float*)d_in[28];
  const float* W1t = (const float*)d_in[29];
  const float* W2t = (const float*)d_in[30];
  const float* b1t = (const float*)d_in[31];
  const float* b2t = (const float*)d_in[32];
  const float* x   = (const float*)d_in[33];
  const float* xr  = (const float*)d_in[34];

  const int N_ = in_sizes[33] / 5;
  const int E_ = in_sizes[0];
  const int* src = ei;
  const int* dst = ei + E_;

  // workspace carve-up (units of 4 bytes)
  float* wsf = (float*)d_ws;
  size_t o = 0;
  auto alloc = [&](size_t cnt) { size_t r = o; o += cnt; return r; };
  const size_t o_cvec  = alloc(128);
  const size_t o_w1    = alloc(128 * MAUG);
  const size_t o_v1    = alloc(96);
  const size_t o_b1s   = alloc(4);
  const size_t o_w2    = alloc(192 * MAUG);
  const size_t o_v2    = alloc(96);
  const size_t o_b2s   = alloc(4);
  const size_t o_h0    = alloc((size_t)N_ * 128);
  const size_t o_h1    = alloc((size_t)N_ * 192);
  const size_t o_h2    = alloc((size_t)N_ * 192);
  const size_t o_hs    = alloc((size_t)N_ * MAUG);
  const size_t o_amax  = alloc((size_t)N_ * 3);  // u32
  const size_t o_denom = alloc((size_t)N_ * 3);
  const size_t o_araw  = alloc((size_t)E_ * 3);
  const size_t o_hs3   = alloc(N_);
  const size_t o_as3   = alloc(N_);
  const size_t o_ad3   = alloc(N_);
  const size_t o_amax3 = alloc(N_);
  const size_t o_den3  = alloc(N_);
  const size_t o_agg3  = alloc(N_);
  (void)ws_size; (void)n_in; (void)out_size;

  const int TB = 256;
  auto blk = [&](long total) { return (int)((total + TB - 1) / TB); };

  // ---- prep ----
  prep_mlp<<<1, 32, 0, stream>>>(W1c, W1t, W1e, wsf + o_cvec);
  prep_layer<<<1, 256, 0, stream>>>(Wsrc1, Wdst1, Wedge1, attS1, attD1, attE1, W2e, b2e,
                                    128, wsf + o_w1, wsf + o_v1, wsf + o_b1s);
  prep_layer<<<1, 256, 0, stream>>>(Wsrc2, Wdst2, Wedge2, attS2, attD2, attE2, W2e, b2e,
                                    192, wsf + o_w2, wsf + o_v2, wsf + o_b2s);

  // ---- node MLP -> h0 (N x 128) ----
  node_mlp<<<blk((long)N_ * 64), TB, 0, stream>>>(x, xr, wsf + o_cvec, b1c, W2c, b2c,
                                                  b1t, W2t, b2t, wsf + o_h0, N_);

  const float* ce_ = wsf + o_cvec + 96;

  // ---- conv1: h0(128) -> h1(192) ----
  gemm_wmma<128, 13><<<(N_ + 127) / 128, 256, 0, stream>>>(wsf + o_h0, wsf + o_w1,
                                                           wsf + o_hs, N_);
  fill_u32<<<blk((long)N_ * 3), TB, 0, stream>>>((unsigned*)(wsf + o_amax), ORD_NEG_INF, N_ * 3);
  fill_u32<<<blk((long)N_ * 3), TB, 0, stream>>>((unsigned*)(wsf + o_denom), 0u, N_ * 3);
  fill_u32<<<blk((long)N_ * 192), TB, 0, stream>>>((unsigned*)(wsf + o_h1), 0u, N_ * 192);
  attn_max<<<blk(E_), TB, 0, stream>>>(src, dst, edge_attr, ce_, b1e, wsf + o_v1, wsf + o_b1s,
                                       wsf + o_hs, wsf + o_araw, (unsigned*)(wsf + o_amax), E_);
  attn_weight<<<blk((long)E_ * 3), TB, 0, stream>>>(dst, wsf + o_araw, (unsigned*)(wsf + o_amax),
                                                    wsf + o_denom, E_ * 3);
  scatter_msg4<<<blk((long)E_ * 48), TB, 0, stream>>>(src, dst, wsf + o_hs, wsf + o_araw,
                                                      wsf + o_h1, E_ * 48);
  finalize_layer<<<blk((long)N_ * 192), TB, 0, stream>>>(wsf + o_h1, wsf + o_denom, bias1,
                                                         N_ * 192);

  // ---- conv2: h1(192) -> h2(192) ----
  gemm_wmma<192, 13><<<(N_ + 127) / 128, 256, 0, stream>>>(wsf + o_h1, wsf + o_w2,
                                                           wsf + o_hs, N_);
  fill_u32<<<blk((long)N_ * 3), TB, 0, stream>>>((unsigned*)(wsf + o_amax), ORD_NEG_INF, N_ * 3);
  fill_u32<<<blk((long)N_ * 3), TB, 0, stream>>>((unsigned*)(wsf + o_denom), 0u, N_ * 3);
  fill_u32<<<blk((long)N_ * 192), TB, 0, stream>>>((unsigned*)(wsf + o_h2), 0u, N_ * 192);
  attn_max<<<blk(E_), TB, 0, stream>>>(src, dst, edge_attr, ce_, b1e, wsf + o_v2, wsf + o_b2s,
                                       wsf + o_hs, wsf + o_araw, (unsigned*)(wsf + o_amax), E_);
  attn_weight<<<blk((long)E_ * 3), TB, 0, stream>>>(dst, wsf + o_araw, (unsigned*)(wsf + o_amax),
                                                    wsf + o_denom, E_ * 3);
  scatter_msg4<<<blk((long)E_ * 48), TB, 0, stream>>>(src, dst, wsf + o_hs, wsf + o_araw,
                                                      wsf + o_h2, E_ * 48);
  finalize_layer<<<blk((long)N_ * 192), TB, 0, stream>>>(wsf + o_h2, wsf + o_denom, bias2,
                                                         N_ * 192);

  // ---- conv3: h2(192) -> scalar per node ----
  alpha3_kernel<<<N_, 256, 0, stream>>>(wsf + o_h2, Wsrc3, Wdst3, attS3, attD3,
                                        wsf + o_hs3, wsf + o_as3, wsf + o_ad3, N_);
  fill_u32<<<blk(N_), TB, 0, stream>>>((unsigned*)(wsf + o_amax3), ORD_NEG_INF, N_);
  fill_u32<<<blk(N_), TB, 0, stream>>>((unsigned*)(wsf + o_den3), 0u, N_);
  fill_u32<<<blk(N_), TB, 0, stream>>>((unsigned*)(wsf + o_agg3), 0u, N_);
  attn_max3<<<blk(E_), TB, 0, stream>>>(src, dst, wsf + o_as3, wsf + o_ad3,
                                        wsf + o_araw, (unsigned*)(wsf + o_amax3), E_);
  attn_wsum3<<<blk(E_), TB, 0, stream>>>(src, dst, wsf + o_araw, (unsigned*)(wsf + o_amax3),
                                         wsf + o_hs3, wsf + o_den3, wsf + o_agg3, E_);
  final3<<<blk(N_), TB, 0, stream>>>(wsf + o_agg3, wsf + o_den3, bias3, x, (float*)d_out, N_);
}